// NeuronClass_nn_56109452755002
// MI455X (gfx1250) — compile-verified
//
#include <hip/hip_runtime.h>
#include <cstdint>
#include <cstddef>

// LIF: T=500, B=1024, N=256 ; s[t,b] = x[t,b,:]·w then sequential scan over t.
#define T_STEPS 500
#define BATCH   1024
#define NFEAT   256
#define ROWS    (T_STEPS * BATCH)   /* 512000 flattened (t,b) rows */
#define NTILES  (ROWS / 16)         /* 32000 tiles of 16 rows      */
#define ROW_DW  260                 /* LDS row stride in dwords: 256 + 4 pad (bank-conflict-free, stride%64==4) */

typedef float        v2f   __attribute__((ext_vector_type(2)));
typedef float        v8f   __attribute__((ext_vector_type(8)));
typedef unsigned int u32x4 __attribute__((ext_vector_type(4)));
typedef int          i32x4 __attribute__((ext_vector_type(4)));
typedef int          i32x8 __attribute__((ext_vector_type(8)));

// Issue one TDM load: 16 rows x 256 f32 from global -> LDS, padding +4 dwords
// after every 256 dwords so LDS rows land at stride ROW_DW.
__device__ __forceinline__ void tdm_load_tile(const float* gsrc, uint32_t lds_off) {
  const uint64_t ga = (uint64_t)(uintptr_t)gsrc;
  u32x4 g0;
  g0[0] = 1u;                                   // count=1, user descriptor, no gather
  g0[1] = lds_off;                              // D#.lds_addr (bytes)
  g0[2] = (uint32_t)ga;                         // global_addr[31:0]
  g0[3] = (uint32_t)((ga >> 32) & 0x01FFFFFFull) | (2u << 30); // addr[56:32] | type=2
  i32x8 g1;
  g1[0] = (int)((2u << 16)                      // data_size = 4B
              | (1u << 20)                      // pad_enable
              | (7u << 22)                      // pad_interval = 256 dwords
              | (3u << 25));                    // pad_amount   = 4 dwords
  g1[1] = (int)(256u << 16);                    // tensor_dim0 = 256  (bits 79:48)
  g1[2] = (int)(16u  << 16);                    // tensor_dim1 = 16   (bits 111:80)
  g1[3] = (int)(256u << 16);                    // tile_dim0   = 256  (bits 127:112)
  g1[4] = 16;                                   // tile_dim1=16, tile_dim2=0
  g1[5] = 256;                                  // tensor_dim0_stride = 256
  g1[6] = 0;                                    // stride0 hi / tensor_dim1_stride lo
  g1[7] = 0;
  i32x4 z4 = {0, 0, 0, 0};
#if __clang_major__ >= 23
  i32x8 z8 = {0, 0, 0, 0, 0, 0, 0, 0};
  __builtin_amdgcn_tensor_load_to_lds(g0, g1, z4, z4, z8, 0);
#else
  __builtin_amdgcn_tensor_load_to_lds(g0, g1, z4, z4, 0);
#endif
}

// Kernel 1: S[r] = x_row[r] · w via V_WMMA_F32_16X16X4_F32 (fp32 precision).
// One wave32 per block; double-buffered TDM tile staging; 64 WMMAs per tile.
// A (16x4) = w chunk broadcast over M; B (4x16) = 16 x-rows => D[m][n]=dot(row n).
__global__ void __launch_bounds__(32)
lif_dot_wmma(const float* __restrict__ x, const float* __restrict__ w,
             float* __restrict__ S) {
  __shared__ float xbuf[2][16 * ROW_DW];   // 2 x 16.25 KB tiles
  __shared__ float wlds[NFEAT];            // 1 KB weight copy

  const int lane = (int)threadIdx.x;       // 0..31
  for (int i = lane; i < NFEAT; i += 32) wlds[i] = w[i];
  __syncthreads();

  const int h = lane >> 4;                 // half-wave: K sub-offset
  const int n = lane & 15;                 // output column / row-in-tile
  const int stride = (int)gridDim.x;
  int tile = (int)blockIdx.x;
  if (tile >= NTILES) return;

  tdm_load_tile(x + (size_t)tile * (16 * NFEAT),
                (uint32_t)(uintptr_t)&xbuf[0][0]);
  int cur = 0;
  for (; tile < NTILES; tile += stride) {
    const int next = tile + stride;
    if (next < NTILES) {
      tdm_load_tile(x + (size_t)next * (16 * NFEAT),
                    (uint32_t)(uintptr_t)&xbuf[cur ^ 1][0]);
      __builtin_amdgcn_s_wait_tensorcnt(1);   // oldest TDM (current tile) done
    } else {
      __builtin_amdgcn_s_wait_tensorcnt(0);
    }
    asm volatile("" ::: "memory");            // don't hoist LDS reads above wait

    const float* xb = &xbuf[cur][n * ROW_DW + 2 * h];
    const float* wb = &wlds[2 * h];
    v8f acc = {};
#pragma unroll
    for (int k = 0; k < NFEAT; k += 4) {
      v2f a = *(const v2f*)(wb + k);          // A[m][2h+r] = w[k+2h+r] (all m)
      v2f b = *(const v2f*)(xb + k);          // B[2h+r][n] = x[row n][k+2h+r]
      acc = __builtin_amdgcn_wmma_f32_16x16x4_f32(false, a, false, b,
                                                  (short)0, acc, false, false);
    }
    if (lane < 16) S[(size_t)tile * 16 + lane] = acc[0];  // coalesced 16-lane store
    cur ^= 1;
  }
}

// Kernel 2: per-batch-lane LIF scan over T. v uses previous z (matches scan order).
__global__ void lif_scan(const float* __restrict__ S, float* __restrict__ out) {
  const int b = (int)(blockIdx.x * blockDim.x + threadIdx.x);
  if (b >= BATCH) return;
  const float alpha = (float)(1.0 - 0.05 / 10.0);   // 0.995
  float v = 0.0f, z = 0.0f;
  float* vout = out;
  float* zout = out + (size_t)T_STEPS * BATCH;
  for (int t = 0; t < T_STEPS; ++t) {
    const float s = S[(size_t)t * BATCH + b];
    v = alpha * v + s - z;                          // V_TH = 1
    z = (v - 1.0f > 0.0f) ? 1.0f : 0.0f;
    vout[(size_t)t * BATCH + b] = v;
    zout[(size_t)t * BATCH + b] = z;
  }
}

extern "C" void kernel_launch(void* const* d_in, const int* in_sizes, int n_in,
                              void* d_out, int out_size, void* d_ws, size_t ws_size,
                              hipStream_t stream) {
  (void)in_sizes; (void)n_in; (void)out_size; (void)ws_size;
  const float* x = (const float*)d_in[0];   // [T,B,N] f32
  const float* w = (const float*)d_in[1];   // [N] f32
  float* out = (float*)d_out;               // v_seq[T,B] ++ z_seq[T,B]
  float* S   = (float*)d_ws;                // [T*B] f32 dot results (2 MB)

  lif_dot_wmma<<<2048, 32, 0, stream>>>(x, w, S);
  lif_scan<<<(BATCH + 255) / 256, 256, 0, stream>>>(S, out);
}